// SubgraphMLP_41205916237956
// MI455X (gfx1250) — compile-verified
//
#include <hip/hip_runtime.h>

// Fused PointNet-style MLP for MI455X (gfx1250, wave32, WMMA).
// One wave32 per (B,N) group of S=20 points. All activations in LDS,
// matmuls via V_WMMA_F32_16X16X4_F32 (fp32 A/B/C, matches fp32 reference
// + LayerNorm statistics). HBM traffic = one read of feats + one write of
// out (~38 MB); weights stay hot in WGP$/L2.
//
// Round-1 refinement: zero-padding of the partial M-tile (rows 20..31) is
// done by pointing invalid lanes at a shared zeroed LDS row (one hoisted
// v_cndmask on the base pointer) instead of per-load predication, so the
// inner loop is pure ds_load_2addr + global_load + v_wmma with no exec
// manipulation.

typedef __attribute__((ext_vector_type(2))) float v2f;
typedef __attribute__((ext_vector_type(8))) float v8f;

#define ROWS   20          // S: points per group
#define LDSTR  133         // LDS row stride (floats); 133 % 64 = 5 -> conflict-free
#define GROUPS_PER_BLOCK 2 // 2 waves per block, one group per wave
#define MAXCH  128         // widest activation (layer-2 input / hidden)

struct KParams {
  const float* p[24];      // per layer L: w1,b1,g1,be1,w2,b2,g2,be2 at p[8L+..]
};

// Y[0:20][0:h] = X[0:20][0:cin] @ W[cin][h] + bias, written to LDS.
// A fragment (16x4 f32): lane L holds row (L&15), k = kt*4 + (L>>4)*2 .. +1
// B fragment (4x16 f32): lane L holds col (L&15), k = kt*4 + (L>>4)*2 .. +1
// D fragment (16x16 f32, v8f): lane L holds col (L&15), rows (L>>4)*8 + i
__device__ __forceinline__ void gemm_bias(const float* __restrict__ W,
                                          const float* __restrict__ bias,
                                          const float* X, float* Y,
                                          const float* zrow,
                                          int cin, int h, int lane) {
  const int n_lo  = lane & 15;
  const int khalf = lane >> 4;
  const int nT = h >> 4;
  const int kT = cin >> 2;

  // Hoisted A-fragment row bases. Rows >= 20 read from the shared zero row,
  // so the inner loop has NO lane-dependent control flow or selects.
  const float* X0 = X + n_lo * LDSTR;                               // rows 0..15
  const float* X1 = (n_lo < 4) ? (X + (16 + n_lo) * LDSTR) : zrow;  // rows 16..31

  for (int nt = 0; nt < nT; ++nt) {
    const int n = (nt << 4) + n_lo;
    v8f acc0 = {};
    v8f acc1 = {};
    for (int kt = 0; kt < kT; ++kt) {
      const int kb = (kt << 2) + (khalf << 1);
      v2f b;
      b.x = W[kb * h + n];
      b.y = W[(kb + 1) * h + n];

      v2f a0;
      a0.x = X0[kb];
      a0.y = X0[kb + 1];
      acc0 = __builtin_amdgcn_wmma_f32_16x16x4_f32(
          false, a0, false, b, (short)0, acc0, false, false);

      v2f a1;
      a1.x = X1[kb];
      a1.y = X1[kb + 1];
      acc1 = __builtin_amdgcn_wmma_f32_16x16x4_f32(
          false, a1, false, b, (short)0, acc1, false, false);
    }
    const float bv = bias[n];
#pragma unroll
    for (int i = 0; i < 8; ++i) {            // M-tile 0: rows 0..15
      const int m = (khalf << 3) + i;
      Y[m * LDSTR + n] = acc0[i] + bv;
    }
    if (khalf == 0) {                        // M-tile 1: only rows 16..19 are real
#pragma unroll
      for (int i = 0; i < 4; ++i)
        Y[(16 + i) * LDSTR + n] = acc1[i] + bv;
    }
  }
}

// Row-wise LayerNorm(+affine)+ReLU over h channels; lanes 0..19 own one row each.
__device__ __forceinline__ void ln_relu(float* Y, const float* __restrict__ g,
                                        const float* __restrict__ be,
                                        int h, int lane) {
  if (lane < ROWS) {
    float* row = Y + lane * LDSTR;
    const float hinv = 1.0f / (float)h;
    float s1 = 0.0f, s2 = 0.0f;
    for (int c = 0; c < h; ++c) { float v = row[c]; s1 += v; s2 += v * v; }
    const float mean = s1 * hinv;
    const float var  = s2 * hinv - mean * mean;
    const float inv  = rsqrtf(var + 1e-5f);
    for (int c = 0; c < h; ++c) {
      float v = (row[c] - mean) * inv * g[c] + be[c];
      row[c] = fmaxf(v, 0.0f);
    }
  }
}

__global__ __launch_bounds__(32 * GROUPS_PER_BLOCK, 1)
void fused_pointnet_mlp(const float* __restrict__ feats, KParams kp,
                        float* __restrict__ out, int ngroups) {
  // Per wave: bufA (current activation, up to 128 cols) + bufB (GEMM scratch)
  __shared__ float smem[2 * GROUPS_PER_BLOCK][ROWS * LDSTR];
  __shared__ float zpad[MAXCH];              // shared zero row for M-tile padding

  const int lane = threadIdx.x & 31;
  const int wave = threadIdx.x >> 5;
  const int g    = blockIdx.x * GROUPS_PER_BLOCK + wave;
  const bool active = (g < ngroups);

  float* bufA = smem[2 * wave + 0];
  float* bufB = smem[2 * wave + 1];

  // Warm WGP$/L2 with the weight matrices (global_prefetch_b8 path).
  if (threadIdx.x == 0) {
#pragma unroll
    for (int L = 0; L < 3; ++L) {
      __builtin_prefetch(kp.p[8 * L + 0], 0, 3);
      __builtin_prefetch(kp.p[8 * L + 4], 0, 3);
    }
  }

  // Zero the pad row once (never written again).
  for (int i = threadIdx.x; i < MAXCH; i += 32 * GROUPS_PER_BLOCK)
    zpad[i] = 0.0f;

  // Stage this group's 20x16 input tile into LDS (contiguous 320 floats).
  if (active) {
    const float* src = feats + (size_t)g * (ROWS * 16);
    for (int i = lane; i < ROWS * 16; i += 32)
      bufA[(i >> 4) * LDSTR + (i & 15)] = src[i];
  }
  __syncthreads();

  int cin = 16;
  for (int L = 0; L < 3; ++L) {
    const int h = 32 << L;                    // 32, 64, 128
    const float* w1  = kp.p[8 * L + 0];
    const float* b1  = kp.p[8 * L + 1];
    const float* g1  = kp.p[8 * L + 2];
    const float* be1 = kp.p[8 * L + 3];
    const float* w2  = kp.p[8 * L + 4];
    const float* b2  = kp.p[8 * L + 5];
    const float* g2  = kp.p[8 * L + 6];
    const float* be2 = kp.p[8 * L + 7];

    gemm_bias(w1, b1, bufA, bufB, zpad, cin, h, lane);   // bufA -> bufB
    __syncthreads();
    ln_relu(bufB, g1, be1, h, lane);
    __syncthreads();
    gemm_bias(w2, b2, bufB, bufA, zpad, h, h, lane);     // bufB -> bufA
    __syncthreads();
    ln_relu(bufA, g2, be2, h, lane);
    __syncthreads();

    if (L < 2) {
      // Column-max over the 20 points; concat [Y | mx] in place (cols h..2h-1).
      for (int c = lane; c < h; c += 32) {
        float mx = 0.0f;                           // post-ReLU values are >= 0
        for (int r = 0; r < ROWS; ++r) mx = fmaxf(mx, bufA[r * LDSTR + c]);
        for (int r = 0; r < ROWS; ++r) bufA[r * LDSTR + h + c] = mx;
      }
      cin = 2 * h;
    } else {
      // Final: max_S(concat(Y, broadcast(max_S Y))) == concat(mx, mx).
      float* og = out + (size_t)g * 256;
      for (int c = lane; c < h; c += 32) {
        float mx = 0.0f;
        for (int r = 0; r < ROWS; ++r) mx = fmaxf(mx, bufA[r * LDSTR + c]);
        if (active) { og[c] = mx; og[128 + c] = mx; }
      }
    }
    __syncthreads();
  }
}

extern "C" void kernel_launch(void* const* d_in, const int* in_sizes, int n_in,
                              void* d_out, int out_size, void* d_ws, size_t ws_size,
                              hipStream_t stream) {
  (void)in_sizes; (void)n_in; (void)d_ws; (void)ws_size;

  const float* feats = (const float*)d_in[0];
  KParams kp;
  for (int i = 0; i < 24; ++i) kp.p[i] = (const float*)d_in[1 + i];
  float* out = (float*)d_out;

  const int ngroups = out_size / 256;                       // 4*4096 = 16384
  const int blocks  = (ngroups + GROUPS_PER_BLOCK - 1) / GROUPS_PER_BLOCK;

  fused_pointnet_mlp<<<blocks, 32 * GROUPS_PER_BLOCK, 0, stream>>>(
      feats, kp, out, ngroups);
}